// HolographicMemory_24902220382478
// MI455X (gfx1250) — compile-verified
//
#include <hip/hip_runtime.h>
#include <math.h>

// ---------------------------------------------------------------------------
// Holographic memory: per-row FFT / gate / IFFT on MI455X (gfx1250).
// B=64 rows, N=131072=2^17 complex points per row.
// Four-step FFT: N = N1*N2, N1=256 (strided pass), N2=512 (contiguous).
// DFT-256 = two radix-16 stages, each a complex 16x16x16 matmul via chained
// V_WMMA_F32_16X16X4_F32. DFT-512 = 2*DFT-256 + radix-2 combine.
// Strided global->LDS tile gathers use the Tensor Data Mover (TDM) with
// hardware LDS row padding; compute waves only touch LDS + WMMA.
// ---------------------------------------------------------------------------

#define NB 64
#define NN 131072          // 2^17
#define NTOT (NB * NN)

typedef __attribute__((ext_vector_type(2))) float v2f;
typedef __attribute__((ext_vector_type(8))) float v8f;
typedef unsigned int v4u __attribute__((ext_vector_type(4)));
typedef int v4i __attribute__((ext_vector_type(4)));
typedef int v8i __attribute__((ext_vector_type(8)));

#if defined(__gfx1250__) && __has_builtin(__builtin_amdgcn_tensor_load_to_lds)
#define HAVE_TDM 1
#else
#define HAVE_TDM 0
#endif
#if __has_include(<hip/amd_detail/amd_gfx1250_TDM.h>)
#define TDM_6ARG 1
#else
#define TDM_6ARG 0
#endif

// --------------------------- WMMA primitive -------------------------------
// D = A(16x4) * B(4x16) + C(16x16), f32. A: 2 VGPRs (lane m=l&15 holds
// K = 4t+2*(l>>4)+{0,1}); B: 2 VGPRs (col=l&15, rows 4t+2*(l>>4)+{0,1});
// C/D: 8 VGPRs, element (r + 8*(l>>4), l&15).
__device__ __forceinline__ v8f wmma4(v2f a, v2f b, v8f c) {
  return __builtin_amdgcn_wmma_f32_16x16x4_f32(
      false, a, false, b, (short)0, c, false, false);
}

// --------------------------- small helpers --------------------------------
__device__ __forceinline__ float2 cw(float turns, float sgn) {
  // e^{sgn * i * 2*pi*turns}
  float s, c;
  __sincosf(6.28318530717958647692f * turns, &s, &c);
  return make_float2(c, sgn * s);
}
__device__ __forceinline__ float2 cmulc(float2 a, float2 b) {
  return make_float2(a.x * b.x - a.y * b.y, a.x * b.y + a.y * b.x);
}

// ----------------------- TDM 2D tile load to LDS --------------------------
// Gathers tile_dim1 rows of tile_dim0 elements (element = 1<<dsz bytes, rows
// stride0 elements apart in memory) into LDS, optionally inserting
// pad_amount+1 DWORDs of LDS padding every 2^(pad_interval+1) DWORDs.
#if HAVE_TDM
__device__ __forceinline__ void tdm_load_2d(unsigned ldsOff, const void* gptr,
                                            unsigned dsz, unsigned tile0,
                                            unsigned tile1,
                                            unsigned long long stride0,
                                            int padInterval, int padAmount) {
  const unsigned long long ga = (unsigned long long)gptr;
  v4u g0;
  g0[0] = 1u;                                           // count=1, user mode
  g0[1] = ldsOff;                                       // lds_addr (bytes)
  g0[2] = (unsigned)ga;                                 // global_addr[31:0]
  g0[3] = (unsigned)((ga >> 32) & 0x01FFFFFFu) | 0x80000000u;  // addr|type=2
  unsigned w0 = (dsz << 16);                            // data_size
  if (padInterval >= 0)
    w0 |= (1u << 20) | ((unsigned)padInterval << 22) | ((unsigned)padAmount << 25);
  const unsigned long long td0 = 0x7FFFFFFFull, td1 = 0x7FFFFFFFull;  // no OOB
  v8i g1;
  g1[0] = (int)w0;
  g1[1] = (int)((td0 & 0xFFFFull) << 16);               // tensor_dim0[15:0]
  g1[2] = (int)(((td0 >> 16) & 0xFFFFull) | ((td1 & 0xFFFFull) << 16));
  g1[3] = (int)(((td1 >> 16) & 0xFFFFull) | ((unsigned long long)tile0 << 16));
  g1[4] = (int)(tile1);                                 // tile_dim2 = 0
  g1[5] = (int)(stride0 & 0xFFFFFFFFull);               // dim0 stride
  g1[6] = (int)((stride0 >> 32) & 0xFFFFull);           // stride0 hi | stride1 lo
  g1[7] = 0;
  const v4i z4 = {0, 0, 0, 0};
#if TDM_6ARG
  const v8i z8 = {0, 0, 0, 0, 0, 0, 0, 0};
  __builtin_amdgcn_tensor_load_to_lds(g0, g1, z4, z4, z8, 0);
#else
  __builtin_amdgcn_tensor_load_to_lds(g0, g1, z4, z4, 0);
#endif
  __builtin_amdgcn_s_wait_tensorcnt(0);
}
#endif

// W16 DFT matrix held as WMMA A-operand chunks (4 K-chunks of 4 columns).
struct W16A {
  v2f wr[4];   // Re(W16)
  v2f wi[4];   // Im(W16)
  v2f wiN[4];  // -Im(W16) (f32 WMMA NEG is C-only, so keep negated copy)
};

__device__ __forceinline__ void buildW16(W16A& W, float sgn) {
  const int lane = threadIdx.x & 31;
  const int m = lane & 15, hi = lane >> 4;
  for (int t = 0; t < 4; ++t) {
    for (int j = 0; j < 2; ++j) {
      const int k = 4 * t + 2 * hi + j;
      const float2 w = cw((float)(m * k) * (1.0f / 16.0f), sgn);
      W.wr[t][j] = w.x;
      W.wi[t][j] = w.y;
      W.wiN[t][j] = -w.y;
    }
  }
}

// Complex 16x16 matmul  C = W16 * M, M via accessor getb(row,col).
template <bool HASIM, typename GetB>
__device__ __forceinline__ void cdft16(const W16A& W, GetB getb,
                                       v8f& Cr, v8f& Ci) {
  const int lane = threadIdx.x & 31;
  const int col = lane & 15, hi = lane >> 4;
  Cr = v8f{0, 0, 0, 0, 0, 0, 0, 0};
  Ci = v8f{0, 0, 0, 0, 0, 0, 0, 0};
  for (int t = 0; t < 4; ++t) {
    const float2 e0 = getb(4 * t + 2 * hi + 0, col);
    const float2 e1 = getb(4 * t + 2 * hi + 1, col);
    v2f br; br[0] = e0.x; br[1] = e1.x;
    Cr = wmma4(W.wr[t], br, Cr);
    Ci = wmma4(W.wi[t], br, Ci);
    if (HASIM) {
      v2f bi; bi[0] = e0.y; bi[1] = e1.y;
      Cr = wmma4(W.wiN[t], bi, Cr);
      Ci = wmma4(W.wr[t], bi, Ci);
    }
  }
}

// Per-lane inner twiddle w256^{k*p} (kernel-invariant; computed once).
__device__ __forceinline__ void buildITW(float2* itw, float sgn) {
  const int lane = threadIdx.x & 31;
  const int col = lane & 15, hi = lane >> 4;
  for (int r = 0; r < 8; ++r)
    itw[r] = cw((float)((r + 8 * hi) * col) * (1.0f / 256.0f), sgn);
}

// DFT-256 of one sequence (accessor gets(j)) by ONE wave.
// Output element (r, lane) = F[(lane&15) + 16*(r + 8*(lane>>4))].
template <bool HASIM, typename GetS>
__device__ __forceinline__ void dft256(const W16A& W, const float2* itw,
                                       float2* Hb, GetS gets,
                                       v8f& Fr, v8f& Fi) {
  const int lane = threadIdx.x & 31;
  const int col = lane & 15, hi = lane >> 4;
  // Stage 1: G[k][p] = sum_q s[p+16q] * w16^{qk}
  v8f Gr, Gi;
  cdft16<HASIM>(W, [&](int q, int p) { return gets(p + 16 * q); }, Gr, Gi);
  for (int r = 0; r < 8; ++r) {
    const int k = r + 8 * hi;
    Hb[k * 17 + col] = cmulc(make_float2(Gr[r], Gi[r]), itw[r]);
  }
  __syncthreads();  // uniform: every wave calls dft256 the same number of times
  // Stage 2: F[k+16m] = sum_p H[k][p] * w16^{pm}
  cdft16<true>(W, [&](int p, int k) { return Hb[k * 17 + p]; }, Fr, Fi);
  __syncthreads();  // Hb safe for reuse
}

// --------------------------- reduction kernels ----------------------------
__global__ void kinit(unsigned* maxbits, float* outTail) {
  if (threadIdx.x == 0) *maxbits = 0u;
  if (threadIdx.x < 65) outTail[threadIdx.x] = (threadIdx.x == 64) ? 1.0f : 0.0f;
}

__global__ void __launch_bounds__(256) kmaxabs(const float* __restrict__ z,
                                               unsigned* maxbits, int n) {
  __shared__ unsigned red[256];
  float m = 0.0f;
  for (int i = blockIdx.x * blockDim.x + threadIdx.x; i < n;
       i += gridDim.x * blockDim.x)
    m = fmaxf(m, fabsf(z[i]));
  red[threadIdx.x] = __float_as_uint(m);  // |z|>=0: uint order == float order
  __syncthreads();
  for (int s = 128; s > 0; s >>= 1) {
    if (threadIdx.x < s) {
      unsigned a = red[threadIdx.x], b = red[threadIdx.x + s];
      red[threadIdx.x] = (a > b) ? a : b;
    }
    __syncthreads();
  }
  if (threadIdx.x == 0) atomicMax(maxbits, red[0]);
}

// --------------------- pass A: strided DFT-256 (forward) ------------------
// For each n2: G[k1][n2] = DFT256_{n1}( x[n2 + 512*n1] ) * wN^{-n2*k1}
// Block: 256 thr (8 waves), 16 consecutive n2, 2 units per wave.
__global__ void __launch_bounds__(256) kfwdA(const float* __restrict__ xin,
                                             float2* __restrict__ mid,
                                             const unsigned* __restrict__ maxbits,
                                             int useScale) {
  __shared__ float tinf[256 * 17];      // [n1][n2l] + 1 DWORD row pad (TDM pad)
  __shared__ float2 Hball[8 * 16 * 17];
  __shared__ float2 stg[256 * 18];      // [k1][n2l] output staging
  const int row = blockIdx.x >> 5;
  const int n2base = (blockIdx.x & 31) * 16;
  const int t = threadIdx.x;
  float scale = 1.0f;
  if (useScale) scale = 1.0f / (__uint_as_float(*maxbits) + 1e-6f);
  const float* xr = xin + (size_t)row * NN;
#if HAVE_TDM
  if (t < 32) {  // wave 0 drives the DMA: 16 x 256 f32 tile, stride 512,
                 // +1 DWORD LDS pad per 16-DWORD row (pad_interval=3)
    tdm_load_2d((unsigned)(size_t)(void*)tinf, xr + n2base,
                /*dsz=*/2, /*tile0=*/16, /*tile1=*/256,
                /*stride0=*/512, /*padI=*/3, /*padA=*/0);
  }
#else
  for (int c = 0; c < 16; ++c) {
    const int n2l = t & 15, n1 = c * 16 + (t >> 4);
    tinf[n1 * 17 + n2l] = xr[(n2base + n2l) + 512 * n1];
  }
#endif
  __syncthreads();
  W16A W;
  buildW16(W, -1.0f);
  float2 itw[8];
  buildITW(itw, -1.0f);
  const int wave = t >> 5, lane = t & 31, col = lane & 15, hi = lane >> 4;
  float2* Hb = Hball + wave * 272;
  for (int uu = 0; uu < 2; ++uu) {
    const int n2l = wave * 2 + uu;
    const int n2 = n2base + n2l;
    v8f Fr, Fi;
    dft256<false>(W, itw, Hb,
                  [&](int j) { return make_float2(tinf[j * 17 + n2l] * scale, 0.0f); },
                  Fr, Fi);
    for (int r = 0; r < 8; ++r) {
      const int k1 = col + 16 * (r + 8 * hi);
      const float2 tw = cw((float)n2 * (float)k1 * (1.0f / 131072.0f), -1.0f);
      stg[k1 * 18 + n2l] = cmulc(make_float2(Fr[r], Fi[r]), tw);
    }
  }
  __syncthreads();
  float2* md = mid + (size_t)row * NN;
  for (int c = 0; c < 16; ++c) {   // 128B contiguous runs
    const int n2l = t & 15, k1 = c * 16 + (t >> 4);
    md[k1 * 512 + n2base + n2l] = stg[k1 * 18 + n2l];
  }
}

// ------------- pass B (fwd) / pass 1' (inv): contiguous DFT-512 -----------
template <int INV>
__global__ void __launch_bounds__(256) kctg(const float2* __restrict__ inb,
                                            float2* __restrict__ outb) {
  __shared__ float2 tin[8 * 512];
  __shared__ float2 Hball[8 * 272];
  __shared__ float2 sout[8 * 512];
  const int row = blockIdx.x >> 5;
  const int k1base = (blockIdx.x & 31) * 8;
  const int t = threadIdx.x;
  const float sgn = INV ? 1.0f : -1.0f;
  const float2* src = inb + (size_t)row * NN + (size_t)k1base * 512;
#if HAVE_TDM
  if (t < 32) {  // 32KB linear copy as a 1D TDM tile
    tdm_load_2d((unsigned)(size_t)(void*)tin, src,
                /*dsz=*/3, /*tile0=*/4096, /*tile1=*/1,
                /*stride0=*/4096, /*padI=*/-1, /*padA=*/0);
  }
#else
  for (int c = 0; c < 16; ++c) tin[c * 256 + t] = src[c * 256 + t];
#endif
  __syncthreads();
  W16A W;
  buildW16(W, sgn);
  float2 itw[8];
  buildITW(itw, sgn);
  const int wave = t >> 5, lane = t & 31, col = lane & 15, hi = lane >> 4;
  float2* Hb = Hball + wave * 272;
  const float2* s = tin + wave * 512;
  // DFT512 via DIT split: E=DFT256(even), O=DFT256(odd), radix-2 combine.
  v8f Er, Ei, Or, Oi;
  dft256<true>(W, itw, Hb, [&](int j) { return s[2 * j]; }, Er, Ei);
  dft256<true>(W, itw, Hb, [&](int j) { return s[2 * j + 1]; }, Or, Oi);
  const int k1 = k1base + wave;
  for (int r = 0; r < 8; ++r) {
    const int k = col + 16 * (r + 8 * hi);          // 0..255
    const float2 w = cw((float)k * (1.0f / 512.0f), sgn);
    const float2 e = make_float2(Er[r], Ei[r]);
    const float2 o = cmulc(make_float2(Or[r], Oi[r]), w);
    float2 x0 = make_float2(e.x + o.x, e.y + o.y);
    float2 x1 = make_float2(e.x - o.x, e.y - o.y);
    if (INV) {
      x0 = cmulc(x0, cw((float)k * (float)k1 * (1.0f / 131072.0f), 1.0f));
      x1 = cmulc(x1, cw((float)(k + 256) * (float)k1 * (1.0f / 131072.0f), 1.0f));
    }
    sout[wave * 512 + k] = x0;
    sout[wave * 512 + 256 + k] = x1;
  }
  __syncthreads();
  float2* dst = outb + (size_t)row * NN + (size_t)k1base * 512;
  for (int c = 0; c < 16; ++c) dst[c * 256 + t] = sout[c * 256 + t];
}

// ----------------- pointwise gating in the (permuted) spectrum ------------
__global__ void __launch_bounds__(256) kcombine(float2* __restrict__ Zb,
                                                float2* __restrict__ Tb,
                                                const float* __restrict__ ctrl) {
  for (int i = blockIdx.x * blockDim.x + threadIdx.x; i < NTOT;
       i += gridDim.x * blockDim.x) {
    const int row = i >> 17;  // NN = 2^17
    const float sg = 1.0f / (1.0f + __expf(-ctrl[row * 3 + 0]));
    const float rg = 1.0f / (1.0f + __expf(-ctrl[row * 3 + 1]));
    const float2 Z = Zb[i], T = Tb[i];
    const float2 NT = make_float2(T.x + sg * Z.x, T.y + sg * Z.y);
    float2 RD = cmulc(NT, make_float2(Z.x, -Z.y));  // NT * conj(Z)
    RD.x *= rg;
    RD.y *= rg;
    Tb[i] = NT;   // spectrum of new_trace
    Zb[i] = RD;   // spectrum of read (retrieve folded in; IFFT linear)
  }
}

// --------------- pass 3' (inv): strided IDFT-256, natural output ----------
// y[n2 + 512*n1] = (1/N) * IDFT256_{k1}( g[k1][n2] )
__global__ void __launch_bounds__(256) kinvB(const float2* __restrict__ gin,
                                             float2* __restrict__ yout) {
  __shared__ float2 tin[256 * 17];      // [k1][n2l] + 1 float2 row pad
  __shared__ float2 Hball[8 * 16 * 17];
  __shared__ float2 stg[256 * 18];      // [n1][n2l]
  const int row = blockIdx.x >> 5;
  const int n2base = (blockIdx.x & 31) * 16;
  const int t = threadIdx.x;
  const float2* g = gin + (size_t)row * NN;
#if HAVE_TDM
  if (t < 32) {  // 16 x 256 float2 tile, stride 512 elems,
                 // +2 DWORDs pad per 32-DWORD row (pad_interval=4, amount=1)
    tdm_load_2d((unsigned)(size_t)(void*)tin, g + n2base,
                /*dsz=*/3, /*tile0=*/16, /*tile1=*/256,
                /*stride0=*/512, /*padI=*/4, /*padA=*/1);
  }
#else
  for (int c = 0; c < 16; ++c) {
    const int n2l = t & 15, k1 = c * 16 + (t >> 4);
    tin[k1 * 17 + n2l] = g[k1 * 512 + n2base + n2l];
  }
#endif
  __syncthreads();
  W16A W;
  buildW16(W, 1.0f);
  float2 itw[8];
  buildITW(itw, 1.0f);
  const int wave = t >> 5, lane = t & 31, col = lane & 15, hi = lane >> 4;
  float2* Hb = Hball + wave * 272;
  const float sc = 1.0f / (float)NN;
  for (int uu = 0; uu < 2; ++uu) {
    const int n2l = wave * 2 + uu;
    v8f Fr, Fi;
    dft256<true>(W, itw, Hb, [&](int j) { return tin[j * 17 + n2l]; }, Fr, Fi);
    for (int r = 0; r < 8; ++r) {
      const int n1 = col + 16 * (r + 8 * hi);
      stg[n1 * 18 + n2l] = make_float2(Fr[r] * sc, Fi[r] * sc);
    }
  }
  __syncthreads();
  float2* y = yout + (size_t)row * NN;
  for (int c = 0; c < 16; ++c) {   // 128B contiguous runs
    const int n2l = t & 15, n1 = c * 16 + (t >> 4);
    y[(n2base + n2l) + 512 * n1] = stg[n1 * 18 + n2l];
  }
}

// ------------------------------- launch -----------------------------------
extern "C" void kernel_launch(void* const* d_in, const int* in_sizes, int n_in,
                              void* d_out, int out_size, void* d_ws,
                              size_t ws_size, hipStream_t stream) {
  const float* z = (const float*)d_in[0];
  const float* trace = (const float*)d_in[1];
  const float* ctrl = (const float*)d_in[2];
  float* out = (float*)d_out;

  char* ws = (char*)d_ws;
  unsigned* maxbits = (unsigned*)ws;
  float2* mid = (float2*)(ws + 256);            // 64 MB (pass intermediate)
  float2* Zb = mid + (size_t)NTOT;              // 64 MB (Z spectrum -> READ)
  float2* Tb = Zb + (size_t)NTOT;               // 64 MB (T spectrum -> NT)

  float2* readOut = (float2*)out;               // complex64 interleaved
  float2* ntrOut = readOut + (size_t)NTOT;
  float* tail = out + (size_t)4 * NTOT;         // dummy_ptr[64] + 1.0f

  kinit<<<1, 128, 0, stream>>>(maxbits, tail);
  kmaxabs<<<512, 256, 0, stream>>>(z, maxbits, NTOT);

  // Forward FFTs (z scaled by 1/(max|z|+1e-6), trace unscaled).
  kfwdA<<<2048, 256, 0, stream>>>(z, mid, maxbits, 1);
  kctg<0><<<2048, 256, 0, stream>>>(mid, Zb);
  kfwdA<<<2048, 256, 0, stream>>>(trace, mid, maxbits, 0);
  kctg<0><<<2048, 256, 0, stream>>>(mid, Tb);

  // Gating / correlation in the spectrum (pointwise, order-agnostic).
  kcombine<<<2048, 256, 0, stream>>>(Zb, Tb, ctrl);

  // new_trace = IFFT(NT)
  kctg<1><<<2048, 256, 0, stream>>>(Tb, mid);
  kinvB<<<2048, 256, 0, stream>>>(mid, ntrOut);
  // read = IFFT(RD)
  kctg<1><<<2048, 256, 0, stream>>>(Zb, mid);
  kinvB<<<2048, 256, 0, stream>>>(mid, readOut);
}